// FEM_69569880261137
// MI455X (gfx1250) — compile-verified
//
#include <hip/hip_runtime.h>
#include <hip/hip_bf16.h>

// ---------------------------------------------------------------------------
// Sizes (fixed by the reference)
// ---------------------------------------------------------------------------
#define Bn   8          // batch
#define Cc   256        // in channels
#define Gg   8          // groups
#define CPG  32         // channels per group
#define Hh   96
#define Ww   96
#define HW   (Hh * Ww)          // 9216
#define NPIX (Bn * HW)          // 73728 pixels total
#define OC   72                 // conv out channels (G*K*K)
#define OCP  80                 // padded to 5 x 16 WMMA N-tiles
#define KK   2304               // GEMM K = 256 * 9
#define NTILES 5                // 80 / 16
#define CHUNK (64 * OCP)        // B-stage chunk: 64 K-rows x 80 cols = 5120 dwords

typedef __attribute__((ext_vector_type(2))) float        v2f;
typedef __attribute__((ext_vector_type(8))) float        v8f;
typedef __attribute__((ext_vector_type(4))) unsigned int v4u;
typedef __attribute__((ext_vector_type(8))) int          v8i;
typedef __attribute__((ext_vector_type(4))) int          v4i;

__device__ __forceinline__ int refl(int v) {
    // reflect pad=1 for size 96:  -1 -> 1,  96 -> 94
    return v < 0 ? -v : (v > (Hh - 1) ? (2 * Hh - 2) - v : v);
}

// ---------------------------------------------------------------------------
// TDM: DMA a contiguous run of `nDwords` dwords from global into LDS.
// Builds a D# descriptor per CDNA5 ISA (1-D tile).
// Tracked with TENSORcnt; caller pairs with s_wait_tensorcnt + barrier.
// ---------------------------------------------------------------------------
#if __has_builtin(__builtin_amdgcn_tensor_load_to_lds)
#define HAVE_TDM 1
__device__ __forceinline__ void tdm_stage(const float* src, float* ldsDst, int nDwords) {
    const unsigned long long ga = (unsigned long long)(uintptr_t)src;
    v4u g0;
    g0.x = 1u;                                          // count=1, user mode
    g0.y = (unsigned)(uintptr_t)ldsDst;                 // lds_addr (byte offset)
    g0.z = (unsigned)ga;                                // global_addr[31:0]
    g0.w = (unsigned)((ga >> 32) & 0x1FFFFFFu) | (2u << 30); // addr[56:32] | type=2
    v8i g1;
    g1[0] = 0x20000;                                    // data_size = 2 (4 bytes)
    g1[1] = (nDwords & 0xFFFF) << 16;                   // tensor_dim0[15:0]
    g1[2] = (nDwords >> 16) | (1 << 16);                // tensor_dim0[31:16] | tensor_dim1=1
    g1[3] = (nDwords & 0xFFFF) << 16;                   // tile_dim0 (fits 16 bits)
    g1[4] = 1;                                          // tile_dim1=1, tile_dim2=0
    g1[5] = nDwords;                                    // tensor_dim0_stride[31:0]
    g1[6] = 0;
    g1[7] = 0;
    v4i g2 = {1, 1, 0, 0};                              // tensor_dim2=1, tensor_dim3=1
    v4i g3 = {0, 0, 0, 0};
#if __clang_major__ >= 23
    v8i g4 = {};
    __builtin_amdgcn_tensor_load_to_lds(g0, g1, g2, g3, g4, 0);
#else
    __builtin_amdgcn_tensor_load_to_lds(g0, g1, g2, g3, 0);
#endif
}
#else
#define HAVE_TDM 0
#endif

// ---------------------------------------------------------------------------
// Pack conv weights (72,256,3,3) -> WT[k = tap*256 + c][oc 0..79], zero pad
// ---------------------------------------------------------------------------
__global__ __launch_bounds__(256) void pack_w_kernel(const float* __restrict__ w,
                                                     float* __restrict__ WT) {
    int idx = blockIdx.x * 256 + threadIdx.x;          // < 2304*80
    if (idx >= KK * OCP) return;
    int oc  = idx % OCP;
    int kk  = idx / OCP;
    int tap = kk / Cc;
    int c   = kk % Cc;
    WT[idx] = (oc < OC) ? w[(size_t)(oc * Cc + c) * 9 + tap] : 0.0f;
}

// ---------------------------------------------------------------------------
// SGE stage 1: gap[b*256+c] = mean over H*W
// ---------------------------------------------------------------------------
__global__ __launch_bounds__(256) void sge_gap_kernel(const float* __restrict__ x,
                                                      float* __restrict__ gap) {
    __shared__ float red[256];
    int bc = blockIdx.x;                                // 0..2047
    const float* p = x + (size_t)bc * HW;
    float s = 0.f;
    for (int i = threadIdx.x; i < HW; i += 256) s += p[i];
    red[threadIdx.x] = s;
    __syncthreads();
    for (int o = 128; o > 0; o >>= 1) {
        if (threadIdx.x < o) red[threadIdx.x] += red[threadIdx.x + o];
        __syncthreads();
    }
    if (threadIdx.x == 0) gap[bc] = red[0] / (float)HW;
}

// ---------------------------------------------------------------------------
// SGE stage 2: one block per (b,g): xn = sum_c xg*gap ; spatial mean / inv(std+eps)
// ---------------------------------------------------------------------------
__global__ __launch_bounds__(256) void sge_xn_kernel(const float* __restrict__ x,
                                                     const float* __restrict__ gap,
                                                     float* __restrict__ xn,
                                                     float* __restrict__ tst) {
    __shared__ float g[CPG];
    __shared__ float r1[256], r2[256];
    int bg = blockIdx.x;                // b*8 + group
    int cbase = bg * CPG;               // == b*256 + g*32 (group-major)
    if (threadIdx.x < CPG) g[threadIdx.x] = gap[cbase + threadIdx.x];
    __syncthreads();
    const float* xb = x + (size_t)cbase * HW;
    float s1 = 0.f, s2 = 0.f;
    for (int i = threadIdx.x; i < HW; i += 256) {
        float a = 0.f;
#pragma unroll 8
        for (int c = 0; c < CPG; ++c) a += xb[(size_t)c * HW + i] * g[c];
        xn[(size_t)bg * HW + i] = a;
        s1 += a;
        s2 += a * a;
    }
    r1[threadIdx.x] = s1;
    r2[threadIdx.x] = s2;
    __syncthreads();
    for (int o = 128; o > 0; o >>= 1) {
        if (threadIdx.x < o) {
            r1[threadIdx.x] += r1[threadIdx.x + o];
            r2[threadIdx.x] += r2[threadIdx.x + o];
        }
        __syncthreads();
    }
    if (threadIdx.x == 0) {
        float n    = (float)HW;
        float mean = r1[0] / n;
        float var  = (r2[0] - r1[0] * r1[0] / n) / (n - 1.0f);   // unbiased (torch .std())
        tst[bg * 2]     = mean;
        tst[bg * 2 + 1] = 1.0f / (sqrtf(var) + 1e-5f);
    }
}

// ---------------------------------------------------------------------------
// SGE stage 3: gate and xe = x * sigmoid(t*w + b)
// ---------------------------------------------------------------------------
__global__ __launch_bounds__(256) void sge_gate_kernel(const float* __restrict__ x,
                                                       const float* __restrict__ xn,
                                                       const float* __restrict__ tst,
                                                       const float* __restrict__ sw,
                                                       const float* __restrict__ sb,
                                                       float* __restrict__ xe) {
    size_t idx = (size_t)blockIdx.x * 256 + threadIdx.x;   // < 8*256*9216
    int i   = (int)(idx % HW);
    int bc  = (int)(idx / HW);          // b*256 + c
    int bg  = bc >> 5;                  // b*8 + c/32
    int gi  = bg & 7;
    float t = (xn[(size_t)bg * HW + i] - tst[bg * 2]) * tst[bg * 2 + 1];
    t = t * sw[gi] + sb[gi];
    float gate = 1.0f / (1.0f + __expf(-t));
    xe[idx] = x[idx] * gate;
}

// ---------------------------------------------------------------------------
// Conv as GEMM with V_WMMA_F32_16X16X4_F32.
//   M = 16 pixels per wave, N = 80 channels (5 tiles), K = 2304 in steps of 4.
//   B (weights) staged into double-buffered LDS by the Tensor Data Mover
//   (wave 0 issues tensor_load_to_lds for chunk s+1 while all 8 waves compute
//   chunk s; sync = s_wait_tensorcnt 0 + block barrier). 36 chunks total.
//   Also accumulates per-channel sum / sumsq for BatchNorm via atomics.
// ---------------------------------------------------------------------------
__global__ __launch_bounds__(256) void conv_wmma_kernel(const float* __restrict__ x,
                                                        const float* __restrict__ WT,
                                                        float* __restrict__ S,
                                                        float* __restrict__ bsum,
                                                        float* __restrict__ bsq) {
    __shared__ float Bs[2][CHUNK];                       // 2 x 20 KB weight stages

    const int tid  = threadIdx.x;
    const int wid  = tid >> 5;
    const int lane = tid & 31;
    const int lm   = lane & 15;
    const int half = lane >> 4;

    const int pixTile = blockIdx.x * 8 + wid;            // 4608 tiles of 16 pixels
    const int pixBase = pixTile * 16;
    const int b   = pixBase / HW;
    const int rem = pixBase - b * HW;
    const int y   = rem / Ww;
    const int x0  = rem - y * Ww;                        // W%16==0 -> same row

    v8f acc[NTILES] = {};                                // 5 x 16x16 f32 accumulators

    const float* xb = x + (size_t)b * Cc * HW;

#if HAVE_TDM
    if (wid == 0) tdm_stage(WT, &Bs[0][0], CHUNK);       // prologue: chunk 0
#endif

    for (int s = 0; s < 36; ++s) {                       // chunk = (tap, 64-ch block)
        const int tap = s >> 2;
        const int c0  = (s & 3) * 64;
        const int dy = tap / 3 - 1, dx = tap % 3 - 1;
        const int yy = refl(y + dy);
        const int xx = refl(x0 + lm + dx);
        const float* abase = xb + yy * Ww + xx;          // + c*HW per channel
        const float* Bcur;

#if HAVE_TDM
        if (wid == 0) __builtin_amdgcn_s_wait_tensorcnt(0);  // chunk s landed
        __syncthreads();                                     // visible to all waves
        if (wid == 0 && s < 35)                              // overlap: DMA chunk s+1
            tdm_stage(WT + (size_t)(s + 1) * CHUNK, &Bs[(s + 1) & 1][0], CHUNK);
        Bcur = &Bs[s & 1][0];
#else
        __syncthreads();
        for (int i = tid; i < CHUNK; i += 256)
            Bs[s & 1][i] = WT[(size_t)s * CHUNK + i];
        __syncthreads();
        Bcur = &Bs[s & 1][0];
#endif
        // warm L2 for the next channel block of the A stream
        if ((s & 3) < 3) __builtin_prefetch(abase + (size_t)(c0 + 64) * HW, 0, 0);

#pragma unroll
        for (int ks = 0; ks < 16; ++ks) {
            const int cl = ks * 4 + 2 * half;            // local K row for this lane
            const int cg = c0 + cl;                      // global input channel
            v2f a;                                       // A 16x4 f32 fragment
            a[0] = abase[(size_t)cg * HW];               // K = 0 / 2
            a[1] = abase[(size_t)(cg + 1) * HW];         // K = 1 / 3
            const float* brow0 = &Bcur[cl * OCP + lm];
            const float* brow1 = &Bcur[(cl + 1) * OCP + lm];
#pragma unroll
            for (int nt = 0; nt < NTILES; ++nt) {
                v2f bf;                                  // B 4x16 f32 fragment
                bf[0] = brow0[nt * 16];
                bf[1] = brow1[nt * 16];
                acc[nt] = __builtin_amdgcn_wmma_f32_16x16x4_f32(
                    false, a, false, bf, (short)0, acc[nt], false, false);
            }
        }
    }

    // Store conv result S[pix][80] and per-channel BN partial sums.
#pragma unroll
    for (int nt = 0; nt < NTILES; ++nt) {
        float s1 = 0.f, s2 = 0.f;
#pragma unroll
        for (int r = 0; r < 8; ++r) {
            const float v = acc[nt][r];
            s1 += v;
            s2 += v * v;
            const int M = r + 8 * half;                  // C/D layout: M = r (+8 hi half)
            S[(size_t)(pixBase + M) * OCP + nt * 16 + lm] = v;
        }
        // lanes l and l^16 hold the same channel -> combine, then one atomic per ch
        s1 += __shfl_xor(s1, 16, 32);
        s2 += __shfl_xor(s2, 16, 32);
        const int ch = nt * 16 + lm;
        if (half == 0 && ch < OC) {
            atomicAdd(&bsum[ch], s1);
            atomicAdd(&bsq[ch], s2);
        }
    }
}

// ---------------------------------------------------------------------------
// Fold BN batch stats into per-channel scale/shift
// ---------------------------------------------------------------------------
__global__ void bn_finalize_kernel(const float* __restrict__ bsum,
                                   const float* __restrict__ bsq,
                                   const float* __restrict__ gamma,
                                   const float* __restrict__ beta,
                                   float* __restrict__ bss) {
    int c = threadIdx.x;
    if (c < OC) {
        const float n    = (float)NPIX;
        const float mean = bsum[c] / n;
        const float var  = bsq[c] / n - mean * mean;     // biased (BN training)
        const float sc   = gamma[c] * rsqrtf(var + 1e-5f);
        bss[c]       = sc;
        bss[OCP + c] = beta[c] - mean * sc;
    }
}

// ---------------------------------------------------------------------------
// BN affine + softmax over the 72 channels, one wave per pixel (in-place on S)
// ---------------------------------------------------------------------------
__global__ __launch_bounds__(256) void bn_softmax_kernel(float* __restrict__ S,
                                                         const float* __restrict__ bss) {
    __shared__ float sc[OC], sh[OC];
    const int tid = threadIdx.x;
    if (tid < OC) { sc[tid] = bss[tid]; sh[tid] = bss[OCP + tid]; }
    __syncthreads();
    const int wid  = tid >> 5;
    const int lane = tid & 31;
    const size_t pix = (size_t)blockIdx.x * 8 + wid;
    float* row = S + pix * OCP;

    float v0 = row[lane]      * sc[lane]      + sh[lane];
    float v1 = row[lane + 32] * sc[lane + 32] + sh[lane + 32];
    float v2 = (lane < 8) ? (row[lane + 64] * sc[lane + 64] + sh[lane + 64]) : -3.4e38f;

    float m = fmaxf(v0, fmaxf(v1, v2));
    for (int o = 16; o >= 1; o >>= 1) m = fmaxf(m, __shfl_xor(m, o, 32));
    float e0 = __expf(v0 - m);
    float e1 = __expf(v1 - m);
    float e2 = (lane < 8) ? __expf(v2 - m) : 0.f;
    float s = e0 + e1 + e2;
    for (int o = 16; o >= 1; o >>= 1) s += __shfl_xor(s, o, 32);
    const float inv = 1.0f / s;
    row[lane]      = e0 * inv;
    row[lane + 32] = e1 * inv;
    if (lane < 8) row[lane + 64] = e2 * inv;
}

// ---------------------------------------------------------------------------
// Final gather: out[b,c,y,x] = sum_tap sigma[pix][g*9+tap] * xe_pad[...]
// ---------------------------------------------------------------------------
__global__ __launch_bounds__(256) void out_kernel(const float* __restrict__ xe,
                                                  const float* __restrict__ sig,
                                                  float* __restrict__ out) {
    size_t idx = (size_t)blockIdx.x * 256 + threadIdx.x;   // < 8*256*96*96
    const int xw = (int)(idx % Ww);
    size_t t = idx / Ww;
    const int yv = (int)(t % Hh);
    t /= Hh;
    const int c = (int)(t % Cc);
    const int b = (int)(t / Cc);
    const int g = c >> 5;

    const size_t pix = (size_t)b * HW + yv * Ww + xw;
    const float* sg = sig + pix * OCP + g * 9;
    const float* xc = xe + (size_t)(b * Cc + c) * HW;

    float acc = 0.f;
#pragma unroll
    for (int tap = 0; tap < 9; ++tap) {
        const int dy = tap / 3 - 1, dx = tap % 3 - 1;
        const int yy = refl(yv + dy), xx = refl(xw + dx);
        acc += sg[tap] * xc[yy * Ww + xx];
    }
    out[idx] = acc;
}

// ---------------------------------------------------------------------------
// Host launcher
// ---------------------------------------------------------------------------
extern "C" void kernel_launch(void* const* d_in, const int* in_sizes, int n_in,
                              void* d_out, int out_size, void* d_ws, size_t ws_size,
                              hipStream_t stream) {
    const float* x   = (const float*)d_in[0];   // (8,256,96,96)
    const float* sw  = (const float*)d_in[1];   // (1,8,1,1)
    const float* sb  = (const float*)d_in[2];   // (1,8,1,1)
    const float* cw  = (const float*)d_in[3];   // (72,256,3,3)
    const float* gam = (const float*)d_in[4];   // (72,)
    const float* bet = (const float*)d_in[5];   // (72,)

    float* ws   = (float*)d_ws;
    float* WT   = ws;                           // 2304*80          = 184320
    float* S    = WT + (size_t)KK * OCP;        // 73728*80         = 5898240
    float* xe   = S + (size_t)NPIX * OCP;       // 8*256*9216       = 18874368
    float* gap  = xe + (size_t)Bn * Cc * HW;    // 2048
    float* xn   = gap + Bn * Cc;                // 64*9216          = 589824
    float* tst  = xn + (size_t)Bn * Gg * HW;    // 128
    float* bsum = tst + 128;                    // 80
    float* bsq  = bsum + OCP;                   // 80
    float* bss  = bsq + OCP;                    // 160

    // zero the BN atomic accumulators (graph-capturable memset node)
    hipMemsetAsync(bsum, 0, 2 * OCP * sizeof(float), stream);

    pack_w_kernel<<<(KK * OCP + 255) / 256, 256, 0, stream>>>(cw, WT);
    sge_gap_kernel<<<Bn * Cc, 256, 0, stream>>>(x, gap);
    sge_xn_kernel<<<Bn * Gg, 256, 0, stream>>>(x, gap, xn, tst);
    sge_gate_kernel<<<(Bn * Cc * HW) / 256, 256, 0, stream>>>(x, xn, tst, sw, sb, xe);
    conv_wmma_kernel<<<NPIX / (16 * 8), 256, 0, stream>>>(x, WT, S, bsum, bsq);
    bn_finalize_kernel<<<1, 128, 0, stream>>>(bsum, bsq, gam, bet, bss);
    bn_softmax_kernel<<<NPIX / 8, 256, 0, stream>>>(S, bss);
    out_kernel<<<(Bn * Cc * HW) / 256, 256, 0, stream>>>(xe, S, (float*)d_out);
}